// Sparsemax_60670708023728
// MI455X (gfx1250) — compile-verified
//
#include <hip/hip_runtime.h>

// Sparsemax over rows of length 2048, fp32.
// One row per wave32; CDNA5 async global->LDS staging; ballot/popc compaction;
// Newton solve for tau on the tiny candidate set (elements > max-1).

#define WPB   4      // waves per block (block = 128 threads)
#define NCOL  2048   // row length
#define VPL   16     // float4 per lane: 16 * 4 * 32 = 2048

__global__ __launch_bounds__(WPB * 32, 1)
void sparsemax_kernel(const float* __restrict__ in, float* __restrict__ out, int rows) {
    __shared__ float4 s_buf[WPB * (NCOL / 4)];   // 8 KB per wave

    const int lane = threadIdx.x & 31;
    const int wave = threadIdx.x >> 5;
    const int row  = blockIdx.x * WPB + wave;    // wave-uniform
    if (row >= rows) return;                     // uniform exit; no barriers used anywhere

    const float* src = in  + (size_t)row * NCOL;
    float*       dst = out + (size_t)row * NCOL;
    float*       cand = reinterpret_cast<float*>(&s_buf[wave * (NCOL / 4)]);

    // ---- Stage row into LDS via CDNA5 async copies (ASYNCcnt path) ----
    // Low 32 bits of the flat pointer to LDS == LDS byte offset (aperture mapping).
    const unsigned lds_base = (unsigned)(size_t)cand;
    const unsigned lbyte = lds_base + (unsigned)lane * 16u;
    const unsigned gbyte = (unsigned)lane * 16u;
    const unsigned long long sbase = (unsigned long long)src;
#pragma unroll
    for (int j = 0; j < VPL; ++j) {
        asm volatile("global_load_async_to_lds_b128 %0, %1, %2"
                     :: "v"(lbyte + (unsigned)(j * 512)),
                        "v"(gbyte + (unsigned)(j * 512)),
                        "s"(sbase)
                     : "memory");
    }
    asm volatile("s_wait_asynccnt 0" ::: "memory");

    // ---- Pull row into registers (ds_load_b128, own wave's region only) ----
    const float4* sv = &s_buf[wave * (NCOL / 4)];
    float4 x[VPL];
#pragma unroll
    for (int j = 0; j < VPL; ++j) x[j] = sv[j * 32 + lane];

    // ---- Row max (per-lane then wave32 butterfly) ----
    float m = -3.402823466e38f;
#pragma unroll
    for (int j = 0; j < VPL; ++j)
        m = fmaxf(m, fmaxf(fmaxf(x[j].x, x[j].y), fmaxf(x[j].z, x[j].w)));
#pragma unroll
    for (int o = 16; o > 0; o >>= 1)
        m = fmaxf(m, __shfl_xor(m, o, 32));

    const float thr = m - 1.0f;   // true tau lies in [max-1, max]

    // ---- Compact candidates (x > thr) into LDS (staging buffer is now free) ----
    const unsigned lanelt = (1u << lane) - 1u;
    unsigned cnt = 0;
#pragma unroll
    for (int j = 0; j < VPL; ++j) {
        const float vv[4] = { x[j].x, x[j].y, x[j].z, x[j].w };
#pragma unroll
        for (int k = 0; k < 4; ++k) {
            const float v = vv[k];
            const bool  p = v > thr;
            const unsigned msk = __builtin_amdgcn_ballot_w32(p);
            if (p) cand[cnt + (unsigned)__popc(msk & lanelt)] = v;
            cnt += (unsigned)__popc(msk);
        }
    }
    // Cross-lane LDS RAW: wave's DS ops are in-order, but make the wait explicit.
    asm volatile("s_wait_dscnt 0" ::: "memory");

    // ---- Newton on candidates: tau' = tau + (sum relu(c - tau) - 1)/count ----
    // f(tau) = sum relu - 1 is convex piecewise-linear, f(thr) >= 0, so Newton
    // from thr increases monotonically and is exact once the active set fixes.
    float tau = thr;
    for (int it = 0; it < 16; ++it) {
        float s = 0.0f, c = 0.0f;
        for (unsigned i = (unsigned)lane; i < cnt; i += 32u) {
            const float d = cand[i] - tau;
            if (d > 0.0f) { s += d; c += 1.0f; }
        }
#pragma unroll
        for (int o = 16; o > 0; o >>= 1) {
            s += __shfl_xor(s, o, 32);
            c += __shfl_xor(c, o, 32);
        }
        const float f  = s - 1.0f;
        const float cc = (c > 0.0f) ? c : 1.0f;
        tau += f / cc;
        if (fabsf(f) <= 1e-7f) break;   // uniform (s,c identical on all lanes)
    }

    // ---- Emit relu(x - tau) from registers, coalesced b128 stores ----
    float4* ov = reinterpret_cast<float4*>(dst);
#pragma unroll
    for (int j = 0; j < VPL; ++j) {
        float4 r;
        r.x = fmaxf(x[j].x - tau, 0.0f);
        r.y = fmaxf(x[j].y - tau, 0.0f);
        r.z = fmaxf(x[j].z - tau, 0.0f);
        r.w = fmaxf(x[j].w - tau, 0.0f);
        ov[j * 32 + lane] = r;
    }
}

extern "C" void kernel_launch(void* const* d_in, const int* in_sizes, int n_in,
                              void* d_out, int out_size, void* d_ws, size_t ws_size,
                              hipStream_t stream) {
    const float* x   = (const float*)d_in[0];
    float*       out = (float*)d_out;
    const int rows = in_sizes[0] / NCOL;          // 8*4096 = 32768
    dim3 block(WPB * 32);
    dim3 grid((rows + WPB - 1) / WPB);
    sparsemax_kernel<<<grid, block, 0, stream>>>(x, out, rows);
}